// VSAETopK_40458591929064
// MI455X (gfx1250) — compile-verified
//
#include <hip/hip_runtime.h>
#include <stdint.h>

// Problem dims (fixed by the reference)
#define BDIM 2048
#define DDIM 2048
#define FDIM 32768
#define KSEL 64

typedef float v2f __attribute__((ext_vector_type(2)));
typedef float v8f __attribute__((ext_vector_type(8)));

#define TILE_M 64
#define TILE_N 128
#define TILE_K 64
#define LDSP   68                      // 64 + 4 dword pad (disjoint bank classes)
#define ABUF   (TILE_M * LDSP)         // 4352 floats
#define BBUF   (TILE_N * LDSP)         // 8704 floats
#define BUFSZ  (ABUF + BBUF)           // 13056 floats per double-buffer stage

// CDNA5 async global->LDS copy: 16B per lane, tracked by ASYNCcnt.
// LDS operand is a VGPR holding the LDS byte address (low 32 bits of the
// generic pointer -- LDS aperture lives in the high word per the ISA).
__device__ __forceinline__ void async_copy_b128(const float* g, const float* l) {
    unsigned lds_addr = (unsigned)(uintptr_t)l;
    asm volatile("global_load_async_to_lds_b128 %0, %1, off"
                 :: "v"(lds_addr), "v"(g)
                 : "memory");
}
__device__ __forceinline__ void wait_async0() {
    asm volatile("s_wait_asynccnt 0" ::: "memory");
}

// ---------------- NT GEMM with f32 WMMA + async double buffering ------------
// C[M,N] = A[M,Kc] * B[N,Kc]^T + bias[N]
// 256 threads = 8 waves; block tile 64x128; each wave owns a 32x32 tile as
// 2x2 V_WMMA_F32_16X16X4_F32 accumulators. fp32 WMMA keeps full precision;
// the kernel is HBM-bound (256MB weight stream), so fp32 rate is free.
__global__ __launch_bounds__(256)
void sae_gemm_nt_kernel(const float* __restrict__ A, const float* __restrict__ Bm,
                        const float* __restrict__ bias, float* __restrict__ C,
                        int M, int N, int Kc) {
    extern __shared__ float lds[];

    const int t    = threadIdx.x;
    const int lane = t & 31;
    const int wave = t >> 5;     // 0..7
    const int wm   = wave & 1;   // M sub-tile (x32)
    const int wn   = wave >> 1;  // N sub-tile (x32)
    const int hl   = lane >> 4;  // half-wave: K pair select (A/B), M+8 (C)
    const int lr   = lane & 15;

    const int m0 = blockIdx.y * TILE_M;
    const int n0 = blockIdx.x * TILE_N;

    auto issue_panel = [&](int buf, int kt) {
        const int base = buf * BUFSZ;    // A panel at base, B panel at base+ABUF
#pragma unroll
        for (int r = 0; r < 4; ++r) {    // A: 1024 float4 slots, 4 per thread
            int s = t + r * 256, row = s >> 4, c4 = s & 15;
            async_copy_b128(A + (size_t)(m0 + row) * Kc + kt + c4 * 4,
                            &lds[base + row * LDSP + c4 * 4]);
        }
#pragma unroll
        for (int r = 0; r < 8; ++r) {    // B: 2048 float4 slots, 8 per thread
            int s = t + r * 256, row = s >> 4, c4 = s & 15;
            async_copy_b128(Bm + (size_t)(n0 + row) * Kc + kt + c4 * 4,
                            &lds[base + ABUF + row * LDSP + c4 * 4]);
        }
    };

    v8f acc[2][2] = {};

    issue_panel(0, 0);
    wait_async0();
    __syncthreads();

    int cur = 0;
    for (int kt = 0; kt < Kc; kt += TILE_K) {
        if (kt + TILE_K < Kc) issue_panel(cur ^ 1, kt + TILE_K);  // prefetch next

        const float* As = lds + cur * BUFSZ;
        const float* Bs = As + ABUF;
#pragma unroll
        for (int kk = 0; kk < TILE_K; kk += 4) {
            const int ka = kk + hl * 2;  // per-lane K pair per 16x4 A layout
            v2f a0 = *(const v2f*)&As[(wm * 32 + 0  + lr) * LDSP + ka];
            v2f a1 = *(const v2f*)&As[(wm * 32 + 16 + lr) * LDSP + ka];
            v2f b0 = *(const v2f*)&Bs[(wn * 32 + 0  + lr) * LDSP + ka];
            v2f b1 = *(const v2f*)&Bs[(wn * 32 + 16 + lr) * LDSP + ka];
            acc[0][0] = __builtin_amdgcn_wmma_f32_16x16x4_f32(false, a0, false, b0,
                          (short)0, acc[0][0], false, false);
            acc[0][1] = __builtin_amdgcn_wmma_f32_16x16x4_f32(false, a0, false, b1,
                          (short)0, acc[0][1], false, false);
            acc[1][0] = __builtin_amdgcn_wmma_f32_16x16x4_f32(false, a1, false, b0,
                          (short)0, acc[1][0], false, false);
            acc[1][1] = __builtin_amdgcn_wmma_f32_16x16x4_f32(false, a1, false, b1,
                          (short)0, acc[1][1], false, false);
        }
        wait_async0();       // next panel landed (had whole compute phase)
        __syncthreads();     // all waves done reading 'cur' before overwrite
        cur ^= 1;
    }

    // Epilogue: fuse bias, restage tile in LDS, store coalesced b128.
    // C frag layout: vgpr r -> M = r + 8*hl, N = lr.
    float* ob = lds;         // reuse buffer 0 (safe after loop-end barrier)
#pragma unroll
    for (int j = 0; j < 2; ++j) {
        const int col = wn * 32 + j * 16 + lr;
        const float bv = bias[n0 + col];
#pragma unroll
        for (int i = 0; i < 2; ++i) {
            const int rbase = wm * 32 + i * 16 + hl * 8;
#pragma unroll
            for (int r = 0; r < 8; ++r)
                ob[(rbase + r) * TILE_N + col] = acc[i][j][r] + bv;
        }
    }
    __syncthreads();
#pragma unroll
    for (int r = 0; r < 8; ++r) {         // 2048 float4 slots, 8 per thread
        int s = t + r * 256, row = s >> 5, c4 = s & 31;
        float4 v = *(float4*)&ob[row * TILE_N + c4 * 4];
        *(float4*)&C[(size_t)(m0 + row) * N + n0 + c4 * 4] = v;
    }
}

// ---------------- Per-row abs top-K via LDS radix select --------------------
// One 256-thread workgroup per row; 128KB row lives in LDS (320KB available).
// Async b128 row load, 4 in-LDS histogram rounds, in-place filtered writeback.
__global__ __launch_bounds__(256)
void sae_topk_kernel(float* __restrict__ S) {
    extern __shared__ float smem[];
    float*    row  = smem;                           // FDIM floats
    unsigned* hist = (unsigned*)(smem + FDIM);       // 256 bins
    unsigned* ctrl = hist + 256;                     // [0]=prefix [1]=want

    const int t = threadIdx.x;
    float* g = S + (size_t)blockIdx.x * FDIM;

    // async row load: 32 x b128 per thread, one wait
#pragma unroll
    for (int i = 0; i < FDIM / 1024; ++i) {
        int s4 = (t + i * 256) * 4;
        async_copy_b128(g + s4, &row[s4]);
    }
    wait_async0();
    __syncthreads();

    unsigned prefix = 0;
    unsigned want = KSEL;
    for (int shift = 24; shift >= 0; shift -= 8) {
        hist[t] = 0;   // blockDim==256
        __syncthreads();
        for (int i = 0; i < FDIM / 256; ++i) {
            unsigned key = __float_as_uint(row[t + i * 256]) & 0x7fffffffu;
            bool ok = (shift == 24) || ((key >> (shift + 8)) == prefix);
            if (ok) atomicAdd(&hist[(key >> shift) & 255u], 1u);
        }
        __syncthreads();
        if (t == 0) {
            unsigned cum = 0;
            int d = 255;
            for (; d > 0; --d) {
                unsigned c = hist[d];
                if (cum + c >= want) break;
                cum += c;
            }
            ctrl[0] = (prefix << 8) | (unsigned)d;
            ctrl[1] = want - cum;
        }
        __syncthreads();
        prefix = ctrl[0];
        want   = ctrl[1];
        __syncthreads();
    }

    const unsigned thresh = prefix;  // 32-bit abs-key of the K-th largest
    for (int i = 0; i < FDIM / 256; ++i) {
        float v = row[t + i * 256];
        unsigned key = __float_as_uint(v) & 0x7fffffffu;
        g[t + i * 256] = (key >= thresh) ? v : 0.0f;
    }
}

// ---------------- launch ----------------------------------------------------
extern "C" void kernel_launch(void* const* d_in, const int* in_sizes, int n_in,
                              void* d_out, int out_size, void* d_ws, size_t ws_size,
                              hipStream_t stream) {
    const float* x     = (const float*)d_in[0];
    const float* W_enc = (const float*)d_in[1];
    const float* b_enc = (const float*)d_in[2];
    const float* W_dec = (const float*)d_in[3];
    const float* b_dec = (const float*)d_in[4];
    float* out = (float*)d_out;
    float* mu  = (float*)d_ws;   // B*F f32 = 256MB scratch; reused in place for S

    dim3 blk(256);
    size_t sh_gemm = (size_t)2 * BUFSZ * sizeof(float);               // 104448 B
    size_t sh_topk = (size_t)FDIM * sizeof(float) + 258 * sizeof(unsigned);

    // 1) mu[B,F] = x @ W_enc^T + b_enc      (HBM-bound: 256MB W_enc stream)
    sae_gemm_nt_kernel<<<dim3(FDIM / TILE_N, BDIM / TILE_M), blk, sh_gemm, stream>>>(
        x, W_enc, b_enc, mu, BDIM, FDIM, DDIM);

    // 2) in-place abs top-64 per row -> dense sparse_features
    sae_topk_kernel<<<dim3(BDIM), blk, sh_topk, stream>>>(mu);

    // 3) out[B,D] = S @ W_dec^T + b_dec     (dense decode beats column-gather)
    sae_gemm_nt_kernel<<<dim3(DDIM / TILE_N, BDIM / TILE_M), blk, sh_gemm, stream>>>(
        mu, W_dec, b_dec, out, BDIM, DDIM, FDIM);
}